// Logalike_70677981823614
// MI455X (gfx1250) — compile-verified
//
#include <hip/hip_runtime.h>
#include <hip/hip_bf16.h>
#include <math.h>

// ---------------------------------------------------------------------------
// Logalike: 512x expm(Q * t_n) of a 128x128 rate matrix + gathered log-lik.
// Compute path: V_WMMA_F32_16X16X4_F32 (fp32 matrix core), per-taxon workgroup.
// ---------------------------------------------------------------------------

typedef __attribute__((ext_vector_type(2))) float v2f;
typedef __attribute__((ext_vector_type(8))) float v8f;

#define SDIM   128            // S (state count), fixed: WMMA tiling specialized
#define SQ2    (SDIM * SDIM)  // 16384 floats = 64 KB
#define TAYLOR_M 6            // Taylor degree
#define NSQUARE  8            // squarings (scale = t / 2^8)

// ---------------------------------------------------------------------------
// Kernel 1: Lorentz distances -> t_n = arccosh(max(-<x_i,x_n>_L, 1+1e-7)) / 2
// ---------------------------------------------------------------------------
__global__ void lorentz_t_kernel(const float* __restrict__ X,
                                 const int* __restrict__ iptr,
                                 float* __restrict__ tvals, int N) {
    int n = blockIdx.x * blockDim.x + threadIdx.x;
    if (n >= N) return;
    int i = *iptr;
    const float* xi = X + (size_t)i * 17;
    const float* xn = X + (size_t)n * 17;
    float inner = -xi[0] * xn[0];
#pragma unroll
    for (int j = 1; j < 17; ++j) inner += xn[j] * xi[j];
    float arg = fmaxf(-inner, 1.0f + 1e-7f);          // rho = 1
    tvals[n] = 0.5f * acoshf(arg);                    // d/2
}

// ---------------------------------------------------------------------------
// 128x128x128 fp32 matmul strip: wave `w` computes rows [16w,16w+16) of A*B,
// A,B resident in LDS. Result kept in 8 accumulator tiles (v8f each).
// Fragment layouts per CDNA5 ISA 7.12.2 (32-bit A 16x4, row-striped B/C).
// ---------------------------------------------------------------------------
__device__ __forceinline__ void mm128_strip(const float* __restrict__ A,
                                            const float* __restrict__ B,
                                            v8f acc[8], int wave, int lane) {
    const v8f zero = {0.f, 0.f, 0.f, 0.f, 0.f, 0.f, 0.f, 0.f};
#pragma unroll
    for (int t = 0; t < 8; ++t) acc[t] = zero;

    const int arow  = wave * 16 + (lane & 15);   // M for this lane's A frag
    const int khalf = (lane >> 4) * 2;           // lanes 16-31 carry K+2,K+3
    const int bcol  = lane & 15;                 // N within a tile

    for (int k = 0; k < SDIM; k += 4) {
        v2f a;
        a.x = A[arow * SDIM + k + khalf];
        a.y = A[arow * SDIM + k + khalf + 1];
#pragma unroll
        for (int t = 0; t < 8; ++t) {
            v2f b;
            int c = t * 16 + bcol;
            b.x = B[(k + khalf)     * SDIM + c];
            b.y = B[(k + khalf + 1) * SDIM + c];
            acc[t] = __builtin_amdgcn_wmma_f32_16x16x4_f32(
                /*neg_a=*/false, a, /*neg_b=*/false, b,
                /*c_mod=*/(short)0, acc[t],
                /*reuse_a=*/false, /*reuse_b=*/false);
        }
    }
}

// Write strip back to LDS: D = acc*scale (+ I on the diagonal if addI).
// C/D layout: lanes 0-15 hold M = vgpr, lanes 16-31 hold M = vgpr+8; N = lane%16.
__device__ __forceinline__ void store_strip(float* __restrict__ D,
                                            const v8f acc[8], int wave, int lane,
                                            float scale, bool addI) {
    const int rbase = wave * 16 + 8 * (lane >> 4);
    const int cl    = lane & 15;
#pragma unroll
    for (int t = 0; t < 8; ++t) {
#pragma unroll
        for (int v = 0; v < 8; ++v) {
            int r = rbase + v;
            int c = t * 16 + cl;
            float val = acc[t][v] * scale;
            if (addI && r == c) val += 1.0f;
            D[r * SDIM + c] = val;
        }
    }
}

// ---------------------------------------------------------------------------
// Kernel 2: per-taxon workgroup. expm(Q*t_n) via Taylor+squaring in LDS,
// then 2048-site gathered log-likelihood reduced to site_ll[n].
// Dynamic LDS: Bm[16384] | Tm[16384] | red[256]   (132096 bytes)
// ---------------------------------------------------------------------------
__global__ __launch_bounds__(256) void expm_loglik_kernel(
        const float* __restrict__ Q, const int* __restrict__ cm,
        const float* __restrict__ tvals, const int* __restrict__ iptr,
        float* __restrict__ site_ll, int sites) {
    extern __shared__ float smem[];
    float* Bm  = smem;            // Q * (t / 2^NSQUARE)
    float* Tm  = smem + SQ2;      // running matrix / final P
    float* red = smem + 2 * SQ2;  // reduction scratch

    const int n    = blockIdx.x;
    const int tid  = threadIdx.x;
    const int wave = tid >> 5;
    const int lane = tid & 31;
    const int i_idx = *iptr;

    const float scale = tvals[n] * (1.0f / (float)(1 << NSQUARE));

    // Build scaled generator B = Q * scale, and T = I + B/M (innermost Horner).
    for (int idx = tid; idx < SQ2; idx += 256) {
        float b = Q[idx] * scale;
        Bm[idx] = b;
        int r = idx >> 7, c = idx & (SDIM - 1);
        Tm[idx] = b * (1.0f / (float)TAYLOR_M) + ((r == c) ? 1.0f : 0.0f);
    }
    __syncthreads();

    v8f acc[8];

    // Horner: T <- I + (B*T)/j, j = M-1 .. 1   (5 matmuls)
    for (int j = TAYLOR_M - 1; j >= 1; --j) {
        mm128_strip(Bm, Tm, acc, wave, lane);
        __syncthreads();                               // all reads of Tm done
        store_strip(Tm, acc, wave, lane, 1.0f / (float)j, /*addI=*/true);
        __syncthreads();
    }

    // Squaring: T <- T*T, 8 times  => T = expm(Q * t_n)
    for (int s = 0; s < NSQUARE; ++s) {
        mm128_strip(Tm, Tm, acc, wave, lane);
        __syncthreads();
        store_strip(Tm, acc, wave, lane, 1.0f, /*addI=*/false);
        __syncthreads();
    }

    // Gathered site log-likelihood from P (= Tm) in LDS.
    const float* P  = Tm;
    const float  pi = 1.0f / (float)SDIM;
    float lsum = 0.0f;
    for (int site = tid; site < sites; site += 256) {
        int si = cm[(size_t)i_idx * sites + site];
        int sj = cm[(size_t)n     * sites + site];
        float cur = pi * P[si] * P[sj];                // row 0 of P
        if ((si == sj) && (si != 0))
            cur += pi * P[si * SDIM + si] * P[si * SDIM + sj];
        lsum += logf(cur);
    }
    red[tid] = lsum;
    __syncthreads();
#pragma unroll
    for (int off = 128; off > 0; off >>= 1) {
        if (tid < off) red[tid] += red[tid + off];
        __syncthreads();
    }
    if (tid == 0) site_ll[n] = (n == i_idx) ? 0.0f : red[0];
}

// ---------------------------------------------------------------------------
// Kernel 3: sum site_ll over taxa, divide by N.
// ---------------------------------------------------------------------------
__global__ void final_reduce_kernel(const float* __restrict__ site_ll,
                                    float* __restrict__ out, int N) {
    __shared__ float red[256];
    int tid = threadIdx.x;
    float s = 0.0f;
    for (int nn = tid; nn < N; nn += 256) s += site_ll[nn];
    red[tid] = s;
    __syncthreads();
#pragma unroll
    for (int off = 128; off > 0; off >>= 1) {
        if (tid < off) red[tid] += red[tid + off];
        __syncthreads();
    }
    if (tid == 0) out[0] = red[0] / (float)N;
}

// ---------------------------------------------------------------------------
extern "C" void kernel_launch(void* const* d_in, const int* in_sizes, int n_in,
                              void* d_out, int out_size, void* d_ws, size_t ws_size,
                              hipStream_t stream) {
    const float* Q  = (const float*)d_in[0];   // 128*128 f32
    const float* X  = (const float*)d_in[1];   // N*17 f32
    const int*   cm = (const int*)d_in[2];     // N*SITES i32
    const int* iptr = (const int*)d_in[3];     // scalar i
    float* out = (float*)d_out;

    const int N     = in_sizes[1] / 17;
    const int sites = in_sizes[2] / N;

    float* tvals   = (float*)d_ws;             // N floats
    float* site_ll = tvals + N;                // N floats

    // t_n
    lorentz_t_kernel<<<(N + 255) / 256, 256, 0, stream>>>(X, iptr, tvals, N);

    // per-taxon expm + gathered log-likelihood
    const size_t lds_bytes = (2 * (size_t)SQ2 + 256) * sizeof(float); // 132096
    static bool attr_done = false;
    if (!attr_done) {
        hipFuncSetAttribute((const void*)expm_loglik_kernel,
                            hipFuncAttributeMaxDynamicSharedMemorySize,
                            (int)lds_bytes);
        attr_done = true;
    }
    expm_loglik_kernel<<<N, 256, lds_bytes, stream>>>(Q, cm, tvals, iptr,
                                                      site_ll, sites);

    // final scalar
    final_reduce_kernel<<<1, 256, 0, stream>>>(site_ll, out, N);
}